// GGNN_1881195675860
// MI455X (gfx1250) — compile-verified
//
#include <hip/hip_runtime.h>
#include <math.h>
#include <stdint.h>

typedef float v2f __attribute__((ext_vector_type(2)));
typedef float v8f __attribute__((ext_vector_type(8)));
typedef unsigned int u32x4 __attribute__((ext_vector_type(4)));
typedef int i32x8 __attribute__((ext_vector_type(8)));
typedef int i32x4 __attribute__((ext_vector_type(4)));

#define D 128
#define LDSW 129   // 129 % 64 == 1 -> conflict-free for both B-fragment patterns
#define WSLAB (D * LDSW * 4)  // 66048 bytes per padded 128x128 weight slab

// ---------------------------------------------------------------------------
// TDM: DMA a contiguous 128x128 fp32 block from global memory into LDS,
// inserting 1 pad DWORD after every 128 DWORDs => LDS row stride 129 floats.
// D# layout per CDNA5 ISA (group0/group1; 2-D tile).
// This toolchain exposes the 6-arg builtin:
//   (u32x4 g0, i32x8 g1, i32x4 g2, i32x4 g3, i32x8 ext, i32 cpol)
// ---------------------------------------------------------------------------
__device__ __forceinline__ void tdm_load_weight_slab(const float* src,
                                                     unsigned lds_byte_off) {
  unsigned long long ga = (unsigned long long)(uintptr_t)src;
  u32x4 g0;
  g0[0] = 1u;                                   // count=1, user descriptor
  g0[1] = lds_byte_off;                         // lds_addr (bytes)
  g0[2] = (unsigned)(ga & 0xFFFFFFFFull);       // global_addr[31:0]
  g0[3] = (unsigned)((ga >> 32) & 0x1FFFFFFull) // global_addr[56:32]
        | (2u << 30);                           // type = 2 ("image")
  i32x8 g1;
  g1[0] = (2 << 16)      // data_size = 4 bytes
        | (1 << 20)      // pad_enable
        | (6 << 22)      // pad_interval: pad after 128 DWORDs
        | (0 << 25);     // pad_amount: 1 DWORD
  g1[1] = (int)(128u << 16);   // tensor_dim0[15:0] = 128  (bits 79:48)
  g1[2] = (int)(128u << 16);   // tensor_dim0 hi=0 | tensor_dim1[15:0]=128
  g1[3] = (int)(128u << 16);   // tensor_dim1 hi=0 | tile_dim0 = 128
  g1[4] = 128;                 // tile_dim1 = 128, tile_dim2 = 0 (2-D)
  g1[5] = 128;                 // tensor_dim0_stride = 128 elements
  g1[6] = 0;
  g1[7] = 0;
  i32x4 gz = {0, 0, 0, 0};
  i32x8 gz8 = {0, 0, 0, 0, 0, 0, 0, 0};
  __builtin_amdgcn_tensor_load_to_lds(g0, g1, gz, gz, gz8, 0);
}

// ---------------------------------------------------------------------------
// x[n,:] = embed_table[node_ids[n]+1, :]   (zero-fill padded rows)
// ---------------------------------------------------------------------------
__global__ void gather_embed_kernel(const int* __restrict__ ids,
                                    const float* __restrict__ table,
                                    float* __restrict__ X, int N, int Npad) {
  int idx = blockIdx.x * blockDim.x + threadIdx.x;   // Npad*32 threads
  int n = idx >> 5;
  int c = (idx & 31) * 4;
  if (n >= Npad) return;
  float4 v = make_float4(0.f, 0.f, 0.f, 0.f);
  if (n < N) {
    int row = ids[n] + 1;
    v = *reinterpret_cast<const float4*>(table + (size_t)row * D + c);
  }
  *reinterpret_cast<float4*>(X + (size_t)n * D + c) = v;
}

// ---------------------------------------------------------------------------
// means[t] = mean_d edge_embed_table[t,d]  (7 rows)
// ---------------------------------------------------------------------------
__global__ void edge_means_kernel(const float* __restrict__ T,
                                  float* __restrict__ means) {
  int t = threadIdx.x;
  if (t < 7) {
    float s = 0.f;
    for (int d = 0; d < D; ++d) s += T[t * D + d];
    means[t] = s * (1.0f / D);
  }
}

__global__ void zero_kernel(float4* __restrict__ p, int n4) {
  int i = blockIdx.x * blockDim.x + threadIdx.x;
  if (i < n4) p[i] = make_float4(0.f, 0.f, 0.f, 0.f);
}

__global__ void zero_small_kernel(float* __restrict__ p, int n) {
  int i = threadIdx.x;
  if (i < n) p[i] = 0.f;
}

// ---------------------------------------------------------------------------
// M = X @ W    (N x 128 @ 128 x 128), W row-major W[k][n], TDM-staged in LDS.
// Block: 256 thr = 8 waves; wave handles 16 rows x 128 cols = 8 WMMA tiles.
// ---------------------------------------------------------------------------
__global__ void __launch_bounds__(256) gemm_m_kernel(
    const float* __restrict__ X, const float* __restrict__ W,
    float* __restrict__ M) {
  extern __shared__ float lds[];                    // [128][LDSW]
  if ((threadIdx.x >> 5) == 0) {
    tdm_load_weight_slab(W, 0u);
    __builtin_amdgcn_s_wait_tensorcnt(0);
  }
  __syncthreads();

  const int lane = threadIdx.x & 31;
  const int wave = threadIdx.x >> 5;
  const int r0 = blockIdx.x * 128 + wave * 16;
  const int mr = lane & 15;                 // A row within strip / B-D column
  const int kh = (lane >> 4) << 1;          // K sub-offset: 0 or 2
  const float* xrow = X + (size_t)(r0 + mr) * D;

  v8f acc[8];
#pragma unroll
  for (int t = 0; t < 8; ++t)
#pragma unroll
    for (int j = 0; j < 8; ++j) acc[t][j] = 0.f;

  for (int k0 = 0; k0 < D; k0 += 4) {
    v2f a = *reinterpret_cast<const v2f*>(xrow + k0 + kh);
#pragma unroll
    for (int t = 0; t < 8; ++t) {
      const int n0 = t * 16;
      v2f b;
      b.x = lds[(k0 + kh) * LDSW + n0 + mr];
      b.y = lds[(k0 + kh + 1) * LDSW + n0 + mr];
      acc[t] = __builtin_amdgcn_wmma_f32_16x16x4_f32(
          false, a, false, b, (short)0, acc[t], false, false);
    }
  }

#pragma unroll
  for (int t = 0; t < 8; ++t) {
    const int n0 = t * 16;
#pragma unroll
    for (int j = 0; j < 8; ++j) {
      int row = r0 + j + ((lane < 16) ? 0 : 8);
      M[(size_t)row * D + n0 + (lane & 15)] = acc[t][j];
    }
  }
}

// ---------------------------------------------------------------------------
// Edge scatter: agg[dst] += m[src] * mean(edge_embed[type])
// ---------------------------------------------------------------------------
__global__ void scatter_kernel(const float* __restrict__ M,
                               const int* __restrict__ edges,
                               const int* __restrict__ etype,
                               const float* __restrict__ means,
                               float* __restrict__ AGG, int E) {
  int idx = blockIdx.x * blockDim.x + threadIdx.x;   // E*32 threads
  int e = idx >> 5;
  if (e >= E) return;
  int c = (idx & 31) * 4;
  int s = edges[e];
  int d = edges[E + e];
  float w = means[etype[e] - 1];
  float4 v = *reinterpret_cast<const float4*>(M + (size_t)s * D + c);
  float* a = AGG + (size_t)d * D + c;
  atomicAdd(a + 0, v.x * w);
  atomicAdd(a + 1, v.y * w);
  atomicAdd(a + 2, v.z * w);
  atomicAdd(a + 3, v.w * w);
}

// ---------------------------------------------------------------------------
// GRU gate, fused dual GEMM:
//   U = AGG @ Wih[gate]^T (+bih),  V = X @ Whh[gate]^T (+bhh)
//   mode 0: Out = sigmoid(U+V) = r        mode 1: Out = sigmoid(U+V) = z
//   mode 2: n = tanh(U + r*V); Out = (1-z)*n + z*x
// Weight slices TDM-staged in LDS as Wi[j][k] / Wh[j][k] (stride LDSW).
// ---------------------------------------------------------------------------
__global__ void __launch_bounds__(256) gru_gate_kernel(
    const float* __restrict__ AGG, const float* __restrict__ X,
    const float* __restrict__ Wih, const float* __restrict__ Whh,
    const float* __restrict__ bih, const float* __restrict__ bhh,
    const float* __restrict__ Rb, const float* __restrict__ Zb,
    float* __restrict__ Out, int mode) {
  extern __shared__ float lds[];
  float* Wi = lds;                    // [128][LDSW]
  float* Wh = lds + D * LDSW;         // [128][LDSW]
  const int gate = (mode == 2) ? 2 : mode;
  if ((threadIdx.x >> 5) == 0) {
    tdm_load_weight_slab(Wih + (size_t)gate * D * D, 0u);
    tdm_load_weight_slab(Whh + (size_t)gate * D * D, (unsigned)WSLAB);
    __builtin_amdgcn_s_wait_tensorcnt(0);
  }
  __syncthreads();

  const int lane = threadIdx.x & 31;
  const int wave = threadIdx.x >> 5;
  const int r0 = blockIdx.x * 128 + wave * 16;
  const int mr = lane & 15;
  const int kh = (lane >> 4) << 1;
  const float* arow = AGG + (size_t)(r0 + mr) * D;
  const float* xrow = X + (size_t)(r0 + mr) * D;

  v8f aU[8], aV[8];
#pragma unroll
  for (int t = 0; t < 8; ++t)
#pragma unroll
    for (int j = 0; j < 8; ++j) { aU[t][j] = 0.f; aV[t][j] = 0.f; }

  for (int k0 = 0; k0 < D; k0 += 4) {
    v2f fa = *reinterpret_cast<const v2f*>(arow + k0 + kh);
    v2f fx = *reinterpret_cast<const v2f*>(xrow + k0 + kh);
#pragma unroll
    for (int t = 0; t < 8; ++t) {
      const int n0 = t * 16;
      v2f bU, bV;
      bU.x = Wi[(n0 + mr) * LDSW + k0 + kh];
      bU.y = Wi[(n0 + mr) * LDSW + k0 + kh + 1];
      bV.x = Wh[(n0 + mr) * LDSW + k0 + kh];
      bV.y = Wh[(n0 + mr) * LDSW + k0 + kh + 1];
      aU[t] = __builtin_amdgcn_wmma_f32_16x16x4_f32(
          false, fa, false, bU, (short)0, aU[t], false, false);
      aV[t] = __builtin_amdgcn_wmma_f32_16x16x4_f32(
          false, fx, false, bV, (short)0, aV[t], false, false);
    }
  }

#pragma unroll
  for (int t = 0; t < 8; ++t) {
    const int col0 = t * 16 + (lane & 15);
#pragma unroll
    for (int j = 0; j < 8; ++j) {
      int row = r0 + j + ((lane < 16) ? 0 : 8);
      size_t o = (size_t)row * D + col0;
      float u = aU[t][j] + bih[gate * D + col0];
      float v = aV[t][j] + bhh[gate * D + col0];
      if (mode != 2) {
        Out[o] = 1.0f / (1.0f + expf(-(u + v)));
      } else {
        float r = Rb[o];
        float z = Zb[o];
        float nv = tanhf(u + r * v);
        Out[o] = (1.0f - z) * nv + z * X[o];
      }
    }
  }
}

// ---------------------------------------------------------------------------
// Pooling P1: t[n] = exp(sigmoid(x[n].gate_w + gate_b)); S += sum t
// (sigmoid in (0,1) -> softmax without max-subtraction is safe)
// ---------------------------------------------------------------------------
__global__ void __launch_bounds__(256) gate_logit_kernel(
    const float* __restrict__ X, const float* __restrict__ gw,
    const float* __restrict__ gb, float* __restrict__ tbuf,
    float* __restrict__ Ssum, int N) {
  const int lane = threadIdx.x & 31;
  const int wave = threadIdx.x >> 5;
  int n = blockIdx.x * 8 + wave;
  float part = 0.f;
  if (n < N) {
    const float* xr = X + (size_t)n * D;
    float s = xr[lane] * gw[lane] + xr[lane + 32] * gw[lane + 32] +
              xr[lane + 64] * gw[lane + 64] + xr[lane + 96] * gw[lane + 96];
    for (int o = 16; o > 0; o >>= 1) s += __shfl_down(s, o, 32);
    if (lane == 0) {
      float g = 1.0f / (1.0f + expf(-(s + gb[0])));
      float tv = expf(g);
      tbuf[n] = tv;
      part = tv;
    }
  }
  __shared__ float red[8];
  if (lane == 0) red[wave] = part;
  __syncthreads();
  if (threadIdx.x == 0) {
    float tot = 0.f;
    for (int w = 0; w < 8; ++w) tot += red[w];
    atomicAdd(Ssum, tot);
  }
}

// P2: accum[d] += sum_n t[n] * x[n,d]
__global__ void __launch_bounds__(128) pool_kernel(
    const float* __restrict__ X, const float* __restrict__ tbuf,
    float* __restrict__ accum, int N) {
  int d = threadIdx.x;
  int start = blockIdx.x * 256;
  int end = start + 256; if (end > N) end = N;
  float local = 0.f;
  for (int n = start; n < end; ++n)
    local += tbuf[n] * X[(size_t)n * D + d];
  atomicAdd(&accum[d], local);
}

// P3: out[d] = accum[d] / S
__global__ void finalize_kernel(const float* __restrict__ accum,
                                const float* __restrict__ Ssum,
                                float* __restrict__ out) {
  int d = threadIdx.x;
  out[d] = accum[d] / Ssum[0];
}

// ---------------------------------------------------------------------------
extern "C" void kernel_launch(void* const* d_in, const int* in_sizes, int n_in,
                              void* d_out, int out_size, void* d_ws,
                              size_t ws_size, hipStream_t stream) {
  const int N = 100000, E = 800000, L = 4;
  const int Npad = 100096;                 // 782 * 128
  const int NBLK = Npad / 128;             // 782 GEMM blocks

  const int*   node_ids = (const int*)d_in[0];
  const int*   edges    = (const int*)d_in[1];
  const int*   etypes   = (const int*)d_in[2];
  const float* embed    = (const float*)d_in[3];
  const float* eembed   = (const float*)d_in[4];
  const float* ggnnW    = (const float*)d_in[5];
  const float* Wih      = (const float*)d_in[6];
  const float* Whh      = (const float*)d_in[7];
  const float* bih      = (const float*)d_in[8];
  const float* bhh      = (const float*)d_in[9];
  const float* gate_w   = (const float*)d_in[10];
  const float* gate_b   = (const float*)d_in[11];
  float* out = (float*)d_out;

  const size_t rowsz = (size_t)Npad * D;
  float* ws    = (float*)d_ws;
  float* xA    = ws;
  float* xB    = xA + rowsz;
  float* Mb    = xB + rowsz;               // message buffer, reused as R
  float* AGG   = Mb + rowsz;
  float* Zb    = AGG + rowsz;
  float* means = Zb + rowsz;               // 7 floats (pad to 8)
  float* tbuf  = means + 8;                // Npad floats
  float* accum = tbuf + Npad;              // 128 + 1 floats

  const size_t lds1 = (size_t)WSLAB;       // 66 KB
  const size_t lds2 = 2 * lds1;            // 132 KB

  gather_embed_kernel<<<Npad / 8, 256, 0, stream>>>(node_ids, embed, xA, N, Npad);
  edge_means_kernel<<<1, 32, 0, stream>>>(eembed, means);

  float* xc = xA;
  float* xn = xB;
  for (int l = 0; l < L; ++l) {
    gemm_m_kernel<<<NBLK, 256, lds1, stream>>>(xc, ggnnW + (size_t)l * D * D, Mb);
    zero_kernel<<<(int)(rowsz / 4 + 255) / 256, 256, 0, stream>>>((float4*)AGG, (int)(rowsz / 4));
    scatter_kernel<<<(E * 32) / 256, 256, 0, stream>>>(Mb, edges, etypes, means, AGG, E);
    // r -> Mb (message buffer is dead after scatter), z -> Zb, combine -> xn
    gru_gate_kernel<<<NBLK, 256, lds2, stream>>>(AGG, xc, Wih, Whh, bih, bhh, Mb, Zb, Mb, 0);
    gru_gate_kernel<<<NBLK, 256, lds2, stream>>>(AGG, xc, Wih, Whh, bih, bhh, Mb, Zb, Zb, 1);
    gru_gate_kernel<<<NBLK, 256, lds2, stream>>>(AGG, xc, Wih, Whh, bih, bhh, Mb, Zb, xn, 2);
    float* tmp = xc; xc = xn; xn = tmp;
  }

  zero_small_kernel<<<1, 256, 0, stream>>>(accum, 129);
  gate_logit_kernel<<<(N + 7) / 8, 256, 0, stream>>>(xc, gate_w, gate_b, tbuf, accum + 128, N);
  pool_kernel<<<(N + 255) / 256, 128, 0, stream>>>(xc, tbuf, accum, N);
  finalize_kernel<<<1, 128, 0, stream>>>(accum, accum + 128, out);
}